// SpectralNetLoss_5892695130791
// MI455X (gfx1250) — compile-verified
//
#include <hip/hip_runtime.h>
#include <hip/hip_bf16.h>

typedef __attribute__((ext_vector_type(2))) float v2f;
typedef __attribute__((ext_vector_type(8))) float v8f;

#define NROWS 8192
#define KDIM  32
#define WAVES_PER_BLOCK 8
#define JTILES_PER_WAVE 8   // each wave handles 8 consecutive 16x16 J-tiles

// ---------------------------------------------------------------------------
// Kernel 1: sq[i] = ||Y[i,:]||^2  (and zero the scalar output accumulator)
// ---------------------------------------------------------------------------
__global__ void spectral_sq_kernel(const float* __restrict__ Y,
                                   float* __restrict__ sq,
                                   float* __restrict__ out) {
    int i = blockIdx.x * blockDim.x + threadIdx.x;
    if (i == 0) out[0] = 0.0f;   // d_out is poisoned; main kernel atomically adds
    if (i < NROWS) {
        const float4* y4 = (const float4*)(Y + (size_t)i * KDIM);
        float s = 0.0f;
#pragma unroll
        for (int c = 0; c < KDIM / 4; ++c) {
            float4 v = y4[c];
            s += v.x * v.x + v.y * v.y + v.z * v.z + v.w * v.w;
        }
        sq[i] = s;
    }
}

// ---------------------------------------------------------------------------
// Kernel 2: fused  sum_ij W_ij * max(sq_i + sq_j - 2*(Y Y^T)_ij, 0)
// One wave32 per 8 consecutive 16x16 J-tiles; gram tile via 8x V_WMMA_F32_16X16X4_F32 (K=32).
// ---------------------------------------------------------------------------
__global__ __launch_bounds__(WAVES_PER_BLOCK * 32)
void spectral_loss_kernel(const float* __restrict__ Y,
                          const float* __restrict__ W,
                          const float* __restrict__ sq,
                          float* __restrict__ out) {
    const int lane  = threadIdx.x & 31;
    const int wave  = threadIdx.x >> 5;
    const int row   = lane & 15;        // M (for A) / N (for B) within tile
    const int hi    = lane >> 4;        // 0: K pair {0,1}; 1: K pair {2,3}
    const int khalf = hi * 2;

    const int nTilesJ      = NROWS / 16;                 // 512
    const int nJGroups     = nTilesJ / JTILES_PER_WAVE;  // 64
    const int unit         = blockIdx.x * WAVES_PER_BLOCK + wave;
    const int iTile        = unit / nJGroups;            // 0..511
    const int jGroup       = unit % nJGroups;            // 0..63

    // ---- A fragments: rows of the I-tile (ISA 32-bit 16x4 A layout) ----
    const int ia = iTile * 16 + row;
    const float* ya = Y + (size_t)ia * KDIM + khalf;     // 8B aligned
    v2f a[8];
#pragma unroll
    for (int c = 0; c < 8; ++c)
        a[c] = *(const v2f*)(ya + 4 * c);

    // sq_i for the 8 C-rows this lane owns: M_v = v + 8*hi
    float sqi[8];
    const float* sqi_base = sq + iTile * 16 + 8 * hi;
#pragma unroll
    for (int v = 0; v < 8; ++v)
        sqi[v] = sqi_base[v];

    float waveAcc = 0.0f;

#pragma unroll 1
    for (int jt = 0; jt < JTILES_PER_WAVE; ++jt) {
        const int jTile = jGroup * JTILES_PER_WAVE + jt;

        // ---- B fragments: B = Y^T of J-tile; layout mirrors A with M->N ----
        const int jb = jTile * 16 + row;
        const float* yb = Y + (size_t)jb * KDIM + khalf;
        v2f b[8];
#pragma unroll
        for (int c = 0; c < 8; ++c)
            b[c] = *(const v2f*)(yb + 4 * c);

        // ---- gram tile: 8 chained f32 WMMAs over K ----
        v8f acc = {0.f, 0.f, 0.f, 0.f, 0.f, 0.f, 0.f, 0.f};
#pragma unroll
        for (int c = 0; c < 8; ++c) {
            acc = __builtin_amdgcn_wmma_f32_16x16x4_f32(
                /*neg_a=*/false, a[c],
                /*neg_b=*/false, b[c],
                /*c_mod=*/(short)0, acc,
                /*reuse_a=*/false, /*reuse_b=*/false);
        }

        const float sqj = sq[jTile * 16 + row];

        // ---- stream W tile, fuse distance + clamp + weighted sum ----
        // C/D layout: VGPR v -> (M = v + 8*hi, N = row)
        const float* wbase =
            W + (size_t)(iTile * 16 + 8 * hi) * NROWS + jTile * 16 + row;
        float partial = 0.0f;
#pragma unroll
        for (int v = 0; v < 8; ++v) {
            float w = wbase[(size_t)v * NROWS];
            float d = sqi[v] + sqj - 2.0f * acc[v];
            d = fmaxf(d, 0.0f);
            partial = fmaf(w, d, partial);
        }
        waveAcc += partial;
    }

    // ---- block reduction (LDS) -> one atomic per block ----
    __shared__ float red[WAVES_PER_BLOCK * 32];
    red[threadIdx.x] = waveAcc;
    __syncthreads();
#pragma unroll
    for (int s = (WAVES_PER_BLOCK * 32) / 2; s > 0; s >>= 1) {
        if ((int)threadIdx.x < s)
            red[threadIdx.x] += red[threadIdx.x + s];
        __syncthreads();
    }
    if (threadIdx.x == 0)
        atomicAdd(out, red[0] * (1.0f / (float)NROWS));
}

// ---------------------------------------------------------------------------
extern "C" void kernel_launch(void* const* d_in, const int* in_sizes, int n_in,
                              void* d_out, int out_size, void* d_ws, size_t ws_size,
                              hipStream_t stream) {
    const float* Y = (const float*)d_in[0];   // [8192, 32] f32
    const float* W = (const float*)d_in[1];   // [8192, 8192] f32
    float* out = (float*)d_out;               // scalar f32
    float* sq  = (float*)d_ws;                // 8192 f32 scratch

    (void)in_sizes; (void)n_in; (void)out_size; (void)ws_size;

    spectral_sq_kernel<<<(NROWS + 255) / 256, 256, 0, stream>>>(Y, sq, out);

    const int nTilesI  = NROWS / 16;                            // 512
    const int nJGroups = (NROWS / 16) / JTILES_PER_WAVE;        // 64
    const int units    = nTilesI * nJGroups;                    // 32768 waves
    const int blocks   = units / WAVES_PER_BLOCK;               // 4096
    spectral_loss_kernel<<<blocks, WAVES_PER_BLOCK * 32, 0, stream>>>(Y, W, sq, out);
}